// KVCache_80212809220520
// MI455X (gfx1250) — compile-verified
//
#include <hip/hip_runtime.h>
#include <hip/hip_bf16.h>

// Problem constants (match reference: B, S, T, HD = 8, 2048, 512, 4096)
constexpr int B  = 8;
constexpr int S  = 2048;   // power of two -> mask/shift addressing
constexpr int T  = 512;
constexpr int HD = 4096;   // floats per row -> 16 KB per row

constexpr int ROW_F4 = HD / 4;          // 1024 x 16B per row
constexpr int BLOCK  = 256;             // 8 waves (wave32)
constexpr int ITERS  = ROW_F4 / BLOCK;  // 4 x B128 per lane per row

static_assert((S & (S - 1)) == 0, "S must be power of two");
static_assert((B * S) == (1 << 14), "shift constants below assume B*S == 2^14");
static_assert(ROW_F4 % BLOCK == 0, "row must tile evenly");

// Native clang vector type: legal for nontemporal builtins AND matches the
// async-LDS builtins' expected v4i pointee. Bit-exact copy, so int4 is fine.
typedef int v4i __attribute__((ext_vector_type(4)));

#if __has_builtin(__builtin_amdgcn_global_load_async_to_lds_b128) && \
    __has_builtin(__builtin_amdgcn_global_store_async_from_lds_b128) && \
    __has_builtin(__builtin_amdgcn_s_wait_asynccnt)
#define USE_ASYNC_LDS 1
typedef __attribute__((address_space(1))) v4i* gv4i_ptr;
typedef __attribute__((address_space(3))) v4i* lv4i_ptr;
// CPol: TH in bits [2:0]; TH_NT = 1. ~1 GiB streamed per call through a
// 192 MB L2 with no intra-call reuse -> non-temporal on both sides.
#define CPOL_NT 1
#endif

// ---------------------------------------------------------------------------
// Gather kernel (placed first so its assembly leads the disasm snippet):
// one block per output row (16 KB). Row source is selected via the inverse
// map (uniform per block -> scalar load). Bulk copy goes through the CDNA5
// async global->LDS->global path (ASYNCcnt), B128 per lane. Lanes store
// exactly the LDS slots they loaded, so the only synchronization needed is
// the per-wave s_wait_asynccnt between the load and store phases.
// ---------------------------------------------------------------------------
__global__ __launch_bounds__(BLOCK) void KVCache_gather_rows_kernel(
    const float* __restrict__ k_cache, const float* __restrict__ v_cache,
    const float* __restrict__ k_val,   const float* __restrict__ v_val,
    const int*   __restrict__ map,     float* __restrict__ out) {

    const int gid    = blockIdx.x;        // [0, 2*B*S)
    const int tensor = gid >> 14;         // / (B*S): 0 = K, 1 = V
    const int r      = gid & (B * S - 1);
    const int b      = r >> 11;           // / S
    const int s      = r & (S - 1);

    const int t = map[s];                 // uniform -> scalar load
    const float* cache = tensor ? v_cache : k_cache;
    const float* val   = tensor ? v_val   : k_val;

    const v4i* __restrict__ src = (t >= 0)
        ? (const v4i*)(val   + ((size_t)b * T + (size_t)t) * HD)
        : (const v4i*)(cache + ((size_t)b * S + (size_t)s) * HD);
    v4i* __restrict__ dst =
        (v4i*)(out + ((size_t)tensor * B * S + (size_t)b * S + (size_t)s) * HD);

#if USE_ASYNC_LDS
    __shared__ v4i smem[ROW_F4];          // 16 KB staging buffer
    #pragma unroll
    for (int i = 0; i < ITERS; ++i) {
        const int idx = i * BLOCK + (int)threadIdx.x;
        __builtin_amdgcn_global_load_async_to_lds_b128(
            (gv4i_ptr)(src + idx), (lv4i_ptr)(&smem[idx]), 0, CPOL_NT);
    }
    __builtin_amdgcn_s_wait_asynccnt(0);  // per-wave: our 4 loads landed in LDS
    #pragma unroll
    for (int i = 0; i < ITERS; ++i) {
        const int idx = i * BLOCK + (int)threadIdx.x;
        __builtin_amdgcn_global_store_async_from_lds_b128(
            (gv4i_ptr)(dst + idx), (lv4i_ptr)(&smem[idx]), 0, CPOL_NT);
    }
    // s_endpgm performs an implicit wait-idle, so outstanding async stores
    // complete (and finish reading LDS) before the wave retires.
#else
    #pragma unroll
    for (int i = 0; i < ITERS; ++i) {
        const int idx = i * BLOCK + (int)threadIdx.x;
        v4i x = __builtin_nontemporal_load(src + idx);
        __builtin_nontemporal_store(x, dst + idx);
    }
#endif
}

// ---------------------------------------------------------------------------
// Build inverse map  map[s] = t  if input_pos[t] == s, else -1.
// Single workgroup; __syncthreads orders init vs scatter. 8 KB in d_ws.
// ---------------------------------------------------------------------------
__global__ void KVCache_build_map_kernel(const int* __restrict__ pos,
                                         int* __restrict__ map) {
    for (int s = threadIdx.x; s < S; s += blockDim.x) map[s] = -1;
    __syncthreads();
    for (int t = threadIdx.x; t < T; t += blockDim.x) {
        int p = pos[t];
        if (p >= 0 && p < S) map[p] = t;
    }
}

// ---------------------------------------------------------------------------
// Host launcher
// ---------------------------------------------------------------------------
extern "C" void kernel_launch(void* const* d_in, const int* in_sizes, int n_in,
                              void* d_out, int out_size, void* d_ws, size_t ws_size,
                              hipStream_t stream) {
    (void)in_sizes; (void)n_in; (void)out_size; (void)ws_size;

    const float* k_cache   = (const float*)d_in[0];
    const float* v_cache   = (const float*)d_in[1];
    const float* k_val     = (const float*)d_in[2];
    const float* v_val     = (const float*)d_in[3];
    const int*   input_pos = (const int*)d_in[4];

    int* map = (int*)d_ws;   // S * sizeof(int) = 8 KB scratch

    KVCache_build_map_kernel<<<1, 1024, 0, stream>>>(input_pos, map);

    const int rows = 2 * B * S;  // 32768 blocks, one 16 KB row each
    KVCache_gather_rows_kernel<<<rows, BLOCK, 0, stream>>>(
        k_cache, v_cache, k_val, v_val, map, (float*)d_out);
}